// GammaNorm2D_8796093022626
// MI455X (gfx1250) — compile-verified
//
#include <hip/hip_runtime.h>
#include <hip/hip_bf16.h>

// GammaNorm2D on MI455X (gfx1250, wave32).
// Structure: (1) streaming per-row reduction (sum, sum-log) with a
// V_WMMA_F32_16X16X4_F32-based wave reduction, (2) fp64 per-row Newton solve
// for (k, theta) + Lanczos 1/Gamma coefficients, (3) streaming elementwise
// regularized-lower-incomplete-gamma evaluation.

typedef __attribute__((ext_vector_type(2))) float v2f;
typedef __attribute__((ext_vector_type(8))) float v8f;

#define ROWS       256
#define ROW_ELEMS  (512 * 512)   // 262144
#define BPR1       8             // reduction blocks per row
#define BPR2       16            // elementwise blocks per row
#define BLK        256           // threads per block (8 wave32)

// ---------------------------------------------------------------------------
// Wave32 sum reduction. Preferred path: one V_WMMA_F32_16X16X4_F32 with
// A = lane partials (VGPR0 -> K0 for lanes 0-15, K2 for lanes 16-31; VGPR1=0)
// and B = all-ones. Then D[m,n] = v[m] + v[m+16] (exact f32), so per-lane
// c[0..7] hold pairsum[0..7] (lanes 0-15) / pairsum[8..15] (lanes 16-31);
// 7 adds + shfl_xor(16) complete the 32-lane sum in every lane.
// ---------------------------------------------------------------------------
__device__ __forceinline__ float wave_reduce32(float v) {
#if __has_builtin(__builtin_amdgcn_wmma_f32_16x16x4_f32)
  v2f a; a.x = v;    a.y = 0.0f;
  v2f b; b.x = 1.0f; b.y = 1.0f;
  v8f c = {};
  c = __builtin_amdgcn_wmma_f32_16x16x4_f32(
      /*neg_a=*/false, a, /*neg_b=*/false, b,
      /*c_mod=*/(short)0, c, /*reuse_a=*/false, /*reuse_b=*/false);
  float h = ((c[0] + c[1]) + (c[2] + c[3])) + ((c[4] + c[5]) + (c[6] + c[7]));
  h += __shfl_xor(h, 16, 32);
  return h;
#else
  #pragma unroll
  for (int m = 16; m >= 1; m >>= 1) v += __shfl_xor(v, m, 32);
  return v;
#endif
}

// ---------------------------------------------------------------------------
// Kernel 1: per-row partial sums of x and log(x + 1e-7).
// Grid: ROWS*BPR1 blocks of 256 threads; each block covers 32768 elements
// (8192 float4, 32 coalesced b128 loads per thread).
// ---------------------------------------------------------------------------
__global__ __launch_bounds__(BLK)
void rowstat_kernel(const float* __restrict__ x, float* __restrict__ partials) {
  const int row = blockIdx.x / BPR1;
  const int blk = blockIdx.x % BPR1;
  const int tid = threadIdx.x;

  const int chunk4 = ROW_ELEMS / BPR1 / 4;  // 8192 float4 per block
  const float4* __restrict__ xr =
      reinterpret_cast<const float4*>(x + (size_t)row * ROW_ELEMS) +
      (size_t)blk * chunk4;

  float s = 0.0f, sl = 0.0f;
  #pragma unroll
  for (int i = 0; i < chunk4 / BLK; ++i) {      // 32 iterations
    const float4 v = xr[tid + i * BLK];
    s  += (v.x + v.y) + (v.z + v.w);
    sl += (__logf(v.x + 1e-7f) + __logf(v.y + 1e-7f)) +
          (__logf(v.z + 1e-7f) + __logf(v.w + 1e-7f));
  }

  const float ws = wave_reduce32(s);
  const float wl = wave_reduce32(sl);

  __shared__ float sSum[BLK / 32];
  __shared__ float sLog[BLK / 32];
  const int wid  = tid >> 5;
  const int lane = tid & 31;
  if (lane == 0) { sSum[wid] = ws; sLog[wid] = wl; }
  __syncthreads();
  if (tid == 0) {
    float a = 0.0f, b = 0.0f;
    #pragma unroll
    for (int w = 0; w < BLK / 32; ++w) { a += sSum[w]; b += sLog[w]; }
    const int pi = row * BPR1 + blk;
    partials[2 * pi + 0] = a;
    partials[2 * pi + 1] = b;
  }
}

// ---------------------------------------------------------------------------
// Kernel 2: per-row ML solve (fp64), 256 threads = 256 rows, one block.
// ---------------------------------------------------------------------------
__device__ double digamma_d(double x) {
  double r = 0.0;
  while (x < 6.0) { r -= 1.0 / x; x += 1.0; }
  const double f = 1.0 / (x * x);
  const double t =
      f * (1.0 / 12.0 -
      f * (1.0 / 120.0 -
      f * (1.0 / 252.0 -
      f * (1.0 / 240.0 -
      f * (1.0 / 132.0 -
      f * (691.0 / 32760.0))))));
  return r + log(x) - 0.5 / x - t;
}

// Matches the reference's _trigamma approximation exactly (same fixed point).
__device__ double trigamma_d(double x) {
  const double z  = x + 1.0;
  const double zz = z * z;
  const double a  = 0.2 - 1.0 / (7.0 * zz);
  const double b  = 1.0 - a / zz;
  const double c  = 1.0 + b / (3.0 * z);
  const double d  = 1.0 + c / (2.0 * z);
  return d / z + 1.0 / (x * x);
}

__device__ double gamma_lanczos(double z) {
  const double P[8] = {676.5203681218851,     -1259.1392167224028,
                       771.3234287776531,     -176.6150291621406,
                       12.507343278686905,    -0.13857109526572012,
                       9.984369578019572e-06,  1.5056327351493116e-07};
  double acc = 0.9999999999998099;
  #pragma unroll
  for (int i = 0; i < 8; ++i) acc += P[i] / (z + (double)(i + 1));
  const double t = z + 7.5;  // z + 8 - 0.5
  return 2.5066282746310002 * pow(t, z + 0.5) * exp(-t) * acc / z;
}

__global__ __launch_bounds__(ROWS)
void solve_kernel(const float* __restrict__ partials, float* __restrict__ params) {
  const int row = threadIdx.x;  // one block of 256 threads

  double sum = 0.0, sumlog = 0.0;
  #pragma unroll
  for (int b = 0; b < BPR1; ++b) {
    sum    += (double)partials[2 * (row * BPR1 + b) + 0];
    sumlog += (double)partials[2 * (row * BPR1 + b) + 1];
  }
  const double N        = (double)ROW_ELEMS;
  const double mean_x   = sum / N + 1e-7;      // mean(x + 1e-7)
  const double mean_log = sumlog / N;          // mean(log(x + 1e-7))
  const double s        = log(mean_x) - mean_log;

  const double s3 = s - 3.0;
  double k = (3.0 - s + sqrt(s3 * s3 + 24.0 * s)) / (12.0 * s) + 1e-7;
  #pragma unroll 1
  for (int it = 0; it < 10; ++it) {
    const double nm = log(k) - digamma_d(k) - s;
    const double dn = 1.0 / k - trigamma_d(k);
    k = k - nm / dn;
  }
  const double th = mean_x / k;

  float* p = params + row * 12;
  p[0] = (float)k;
  p[1] = (float)(1.0 / th);
  #pragma unroll 1
  for (int j = 0; j < 8; ++j)
    p[2 + j] = (float)(1.0 / gamma_lanczos(k + (double)(j + 1)));
}

// ---------------------------------------------------------------------------
// Kernel 3: elementwise  out = z^k * e^-z * sum_j c_j z^j,  z = x * (1/theta).
// (Gamma(k) cancels analytically between 1/Gamma(k) and L.)
// ---------------------------------------------------------------------------
__device__ __forceinline__ float gamma_cdf_eval(float xv, float k, float inv_th,
                                                const float* __restrict__ c) {
  const float z = xv * inv_th;
  const float e = __expf(fmaf(k, __logf(z), -z));  // z^k * e^-z
  float poly = c[7];
  #pragma unroll
  for (int j = 6; j >= 0; --j) poly = fmaf(poly, z, c[j]);
  return e * poly;
}

__global__ __launch_bounds__(BLK)
void apply_kernel(const float* __restrict__ x, const float* __restrict__ params,
                  float* __restrict__ out) {
  const int row = blockIdx.x / BPR2;
  const int blk = blockIdx.x % BPR2;
  const int tid = threadIdx.x;

  const float* __restrict__ p = params + row * 12;
  const float kk     = p[0];
  const float inv_th = p[1];
  float c[8];
  #pragma unroll
  for (int j = 0; j < 8; ++j) c[j] = p[2 + j];

  const int chunk4 = ROW_ELEMS / BPR2 / 4;  // 4096 float4 per block
  const float4* __restrict__ xr =
      reinterpret_cast<const float4*>(x + (size_t)row * ROW_ELEMS) +
      (size_t)blk * chunk4;
  float4* __restrict__ outr =
      reinterpret_cast<float4*>(out + (size_t)row * ROW_ELEMS) +
      (size_t)blk * chunk4;

  #pragma unroll
  for (int i = 0; i < chunk4 / BLK; ++i) {    // 16 iterations
    const float4 v = xr[tid + i * BLK];
    float4 r;
    r.x = gamma_cdf_eval(v.x, kk, inv_th, c);
    r.y = gamma_cdf_eval(v.y, kk, inv_th, c);
    r.z = gamma_cdf_eval(v.z, kk, inv_th, c);
    r.w = gamma_cdf_eval(v.w, kk, inv_th, c);
    outr[tid + i * BLK] = r;
  }
}

// ---------------------------------------------------------------------------
extern "C" void kernel_launch(void* const* d_in, const int* in_sizes, int n_in,
                              void* d_out, int out_size, void* d_ws, size_t ws_size,
                              hipStream_t stream) {
  const float* x   = (const float*)d_in[0];
  float*       out = (float*)d_out;
  float*       ws  = (float*)d_ws;

  float* partials = ws;                       // 2 * ROWS * BPR1 = 4096 floats
  float* params   = ws + 2 * ROWS * BPR1;     // ROWS * 12       = 3072 floats

  rowstat_kernel<<<ROWS * BPR1, BLK, 0, stream>>>(x, partials);
  solve_kernel  <<<1,           ROWS, 0, stream>>>(partials, params);
  apply_kernel  <<<ROWS * BPR2, BLK, 0, stream>>>(x, params, out);
}